// DQNNModel_8315056685298
// MI455X (gfx1250) — compile-verified
//
#include <hip/hip_runtime.h>
#include <hip/hip_bf16.h>

#define NQ   12
#define NL   8
#define SEQL 64
#define BATCH 2048
#define DIM  4096   // 2^NQ

typedef float v2f __attribute__((ext_vector_type(2)));
typedef float v8f __attribute__((ext_vector_type(8)));

// D = A(16x4) * B(4x16) + C, f32 WMMA (CDNA5 V_WMMA_F32_16X16X4_F32)
static __device__ __forceinline__ v8f wmma4(v2f a, v2f b, v8f c) {
  return __builtin_amdgcn_wmma_f32_16x16x4_f32(false, a, false, b, (short)0, c, false, false);
}

// ---------------------------------------------------------------------------
// Kernel 0: zero-pad W_in [12][64] -> Wp [16][64] so the projection kernel is
// fully uniform (no EXEC churn around WMMA fragment loads).
// ---------------------------------------------------------------------------
__global__ __launch_bounds__(1024) void pad_kernel(const float* __restrict__ Win,
                                                   float* __restrict__ Wp) {
  const int t = threadIdx.x;           // 0..1023 = 16*64
  const int n = t >> 6, k = t & 63;
  Wp[t] = (n < NQ) ? Win[n * SEQL + k] : 0.0f;
}

// ---------------------------------------------------------------------------
// Kernel 1: build fused per-layer 4-qubit group gates.
// gates[layer][group][ri][16][16]  (float), group 0 = wires 0-3 (MSB side)
// ---------------------------------------------------------------------------
__global__ __launch_bounds__(256) void gate_kernel(const float* __restrict__ qw,
                                                   float* __restrict__ gates) {
  __shared__ float Ur[4][2][2], Ui[4][2][2];
  const int layer = blockIdx.x / 3, grp = blockIdx.x % 3;
  const int tid = threadIdx.x;
  if (tid < 4) {
    const int w = grp * 4 + tid;
    const float* p = qw + (w * NL + layer) * 3;
    const float a = 0.5f * p[0], b = 0.5f * p[1], g = 0.5f * p[2];
    const float ca = cosf(a), sa = sinf(a);
    const float cb = cosf(b), sb = sinf(b);
    const float cg = cosf(g), sg = sinf(g);
    // fused U = RZ(gamma) RY(beta) RX(alpha); ez = e^{-i g} = cg - i sg
    const float m00r = cb * ca,    m00i =  sb * sa;
    const float m01r = -(sb * ca), m01i = -(cb * sa);
    const float m10r = sb * ca,    m10i = -(cb * sa);
    const float m11r = cb * ca,    m11i = -(sb * sa);
    Ur[tid][0][0] = cg * m00r + sg * m00i;  Ui[tid][0][0] = cg * m00i - sg * m00r;
    Ur[tid][0][1] = cg * m01r + sg * m01i;  Ui[tid][0][1] = cg * m01i - sg * m01r;
    Ur[tid][1][0] = cg * m10r - sg * m10i;  Ui[tid][1][0] = cg * m10i + sg * m10r;
    Ur[tid][1][1] = cg * m11r - sg * m11i;  Ui[tid][1][1] = cg * m11i + sg * m11r;
  }
  __syncthreads();
  // G = U0 (x) U1 (x) U2 (x) U3 ; i = i0*8 + i1*4 + i2*2 + i3
  const int i = tid >> 4, j = tid & 15;
  float pr = 1.0f, pi = 0.0f;
#pragma unroll
  for (int q = 0; q < 4; ++q) {
    const int iq = (i >> (3 - q)) & 1, jq = (j >> (3 - q)) & 1;
    const float ur = Ur[q][iq][jq], ui = Ui[q][iq][jq];
    const float nr = pr * ur - pi * ui;
    const float ni = pr * ui + pi * ur;
    pr = nr; pi = ni;
  }
  float* g0 = gates + (size_t)((layer * 3 + grp) * 2) * 256;
  g0[i * 16 + j]        = pr;
  g0[256 + i * 16 + j]  = pi;
}

// ---------------------------------------------------------------------------
// Kernel 2: input projection x = inputs @ Wp^T via f32 WMMA (branch-free).
// One wave per 16-row tile of the batch; N = 16 (padded); K = 64.
// ---------------------------------------------------------------------------
__global__ __launch_bounds__(32) void proj_kernel(const float* __restrict__ inp,
                                                  const float* __restrict__ Wp,
                                                  float* __restrict__ xbuf) {
  const int bm = blockIdx.x;           // 0..127 (2048/16)
  const int lane = threadIdx.x & 31;
  const int lr = lane & 15, hi = lane >> 4;
  const float* ap = inp + (size_t)(bm * 16 + lr) * SEQL;
  const float* bp = Wp  + (size_t)lr * SEQL;     // B[k][n] = Wp[n][k]
  v8f d = {};
#pragma unroll
  for (int s = 0; s < 16; ++s) {
    const int kb = 4 * s + 2 * hi;
    v2f a, b;
    a.x = ap[kb]; a.y = ap[kb + 1];
    b.x = bp[kb]; b.y = bp[kb + 1];
    d = wmma4(a, b, d);
  }
#pragma unroll
  for (int r = 0; r < 8; ++r) {
    const int m = r + 8 * hi;
    xbuf[(size_t)(bm * 16 + m) * 16 + lr] = d[r];
  }
}

// ---------------------------------------------------------------------------
// Kernel 3: full circuit, one workgroup (8 waves) per batch element.
// State (4096 complex) lives in LDS; three WMMA passes + one CNOT-ring
// permutation per layer; PauliZ readout folded with W_out.
// ---------------------------------------------------------------------------
__global__ __launch_bounds__(256) void circuit_kernel(const float* __restrict__ xbuf,
                                                      const float* __restrict__ gates,
                                                      const float* __restrict__ Wout,
                                                      const float* __restrict__ bout,
                                                      float* __restrict__ out) {
  __shared__ float sR[DIM];
  __shared__ float sI[DIM];
  __shared__ float cw[NQ], sw[NQ];
  __shared__ float red[256];

  const int b = blockIdx.x;
  const int tid = threadIdx.x;
  const int lane = tid & 31, wave = tid >> 5;
  const int lr = lane & 15, hi = lane >> 4;

  // --- product-state encoding angles ---
  if (tid < NQ) {
    const float th = 1.5707963267948966f * xbuf[(size_t)b * 16 + tid];
    cw[tid] = cosf(th);
    sw[tid] = sinf(th);
  }
  __syncthreads();
#pragma unroll
  for (int j = 0; j < 16; ++j) {
    const int y = tid + 256 * j;
    float p = 1.0f;
#pragma unroll
    for (int w = 0; w < NQ; ++w)
      p *= ((y >> (11 - w)) & 1) ? sw[w] : cw[w];
    sR[y] = p;
    sI[y] = 0.0f;
  }
  __syncthreads();

  for (int layer = 0; layer < NL; ++layer) {
    const float* gl = gates + (size_t)layer * 3 * 2 * 256;

    // ---------------- pass A: wires 0-3 (bits 11-8), D = G_A @ S[16][256] ----
    {
      const float* gr = gl;
      const float* gi = gr + 256;
      v2f ar[4], ai[4], an[4];
#pragma unroll
      for (int s = 0; s < 4; ++s) {
        const int kb = 4 * s + 2 * hi;
        ar[s].x = gr[lr * 16 + kb]; ar[s].y = gr[lr * 16 + kb + 1];
        ai[s].x = gi[lr * 16 + kb]; ai[s].y = gi[lr * 16 + kb + 1];
        an[s] = -ai[s];
      }
#pragma unroll
      for (int t = 0; t < 2; ++t) {
        const int col = (wave + 8 * t) * 16 + lr;
        v2f br[4], bi[4];
#pragma unroll
        for (int s = 0; s < 4; ++s) {
          const int kb = 4 * s + 2 * hi;
          br[s].x = sR[(kb << 8) + col]; br[s].y = sR[((kb + 1) << 8) + col];
          bi[s].x = sI[(kb << 8) + col]; bi[s].y = sI[((kb + 1) << 8) + col];
        }
        v8f dr = {}, di = {};
#pragma unroll
        for (int s = 0; s < 4; ++s) dr = wmma4(ar[s], br[s], dr);
#pragma unroll
        for (int s = 0; s < 4; ++s) dr = wmma4(an[s], bi[s], dr);
#pragma unroll
        for (int s = 0; s < 4; ++s) di = wmma4(ar[s], bi[s], di);
#pragma unroll
        for (int s = 0; s < 4; ++s) di = wmma4(ai[s], br[s], di);
#pragma unroll
        for (int r = 0; r < 8; ++r) {
          const int m = r + 8 * hi;
          sR[(m << 8) + col] = dr[r];
          sI[(m << 8) + col] = di[r];
        }
      }
    }
    __syncthreads();

    // ---------------- pass B: wires 4-7 (bits 7-4), per h-slice D = G_B @ S_h
    {
      const float* gr = gl + 512;
      const float* gi = gr + 256;
      v2f ar[4], ai[4], an[4];
#pragma unroll
      for (int s = 0; s < 4; ++s) {
        const int kb = 4 * s + 2 * hi;
        ar[s].x = gr[lr * 16 + kb]; ar[s].y = gr[lr * 16 + kb + 1];
        ai[s].x = gi[lr * 16 + kb]; ai[s].y = gi[lr * 16 + kb + 1];
        an[s] = -ai[s];
      }
#pragma unroll
      for (int t = 0; t < 2; ++t) {
        const int hbase = (wave + 8 * t) << 8;
        v2f br[4], bi[4];
#pragma unroll
        for (int s = 0; s < 4; ++s) {
          const int kb = 4 * s + 2 * hi;
          br[s].x = sR[hbase + (kb << 4) + lr];
          br[s].y = sR[hbase + ((kb + 1) << 4) + lr];
          bi[s].x = sI[hbase + (kb << 4) + lr];
          bi[s].y = sI[hbase + ((kb + 1) << 4) + lr];
        }
        v8f dr = {}, di = {};
#pragma unroll
        for (int s = 0; s < 4; ++s) dr = wmma4(ar[s], br[s], dr);
#pragma unroll
        for (int s = 0; s < 4; ++s) dr = wmma4(an[s], bi[s], dr);
#pragma unroll
        for (int s = 0; s < 4; ++s) di = wmma4(ar[s], bi[s], di);
#pragma unroll
        for (int s = 0; s < 4; ++s) di = wmma4(ai[s], br[s], di);
#pragma unroll
        for (int r = 0; r < 8; ++r) {
          const int m = r + 8 * hi;
          sR[hbase + (m << 4) + lr] = dr[r];
          sI[hbase + (m << 4) + lr] = di[r];
        }
      }
    }
    __syncthreads();

    // ---------------- pass C: wires 8-11 (bits 3-0), D = S[256][16] @ G_C^T --
    {
      const float* gr = gl + 1024;
      const float* gi = gr + 256;
      v2f bgr[4], bgi[4], bgn[4];   // B[k][n] = G_C[n][k]
#pragma unroll
      for (int s = 0; s < 4; ++s) {
        const int kb = 4 * s + 2 * hi;
        bgr[s].x = gr[lr * 16 + kb]; bgr[s].y = gr[lr * 16 + kb + 1];
        bgi[s].x = gi[lr * 16 + kb]; bgi[s].y = gi[lr * 16 + kb + 1];
        bgn[s] = -bgi[s];
      }
#pragma unroll
      for (int t = 0; t < 2; ++t) {
        const int rt = wave + 8 * t;
        v2f asr[4], asi[4];         // A[m][k] = S[(rt*16+m)<<4 | k]
#pragma unroll
        for (int s = 0; s < 4; ++s) {
          const int kb = 4 * s + 2 * hi;
          const int base = ((rt * 16 + lr) << 4) + kb;
          asr[s].x = sR[base]; asr[s].y = sR[base + 1];
          asi[s].x = sI[base]; asi[s].y = sI[base + 1];
        }
        v8f dr = {}, di = {};
#pragma unroll
        for (int s = 0; s < 4; ++s) dr = wmma4(asr[s], bgr[s], dr);
#pragma unroll
        for (int s = 0; s < 4; ++s) dr = wmma4(asi[s], bgn[s], dr);
#pragma unroll
        for (int s = 0; s < 4; ++s) di = wmma4(asr[s], bgi[s], di);
#pragma unroll
        for (int s = 0; s < 4; ++s) di = wmma4(asi[s], bgr[s], di);
#pragma unroll
        for (int r = 0; r < 8; ++r) {
          const int m = r + 8 * hi;
          const int idx = ((rt * 16 + m) << 4) + lr;
          sR[idx] = dr[r];
          sI[idx] = di[r];
        }
      }
    }
    __syncthreads();

    // ---------------- CNOT ring: one GF(2)-linear gather --------------------
    // src bits: s0=y0^y11, s1=y1^y0^y11, s_k=y_k^y_{k-1} (k>=2)
    float tr[16], ti[16];
#pragma unroll
    for (int j = 0; j < 16; ++j) {
      const int y = tid + 256 * j;
      const int src = ((y ^ (y >> 1)) & 0x3FF)
                    | ((((y >> 11) ^ y) & 1) << 11)
                    | ((((y >> 10) ^ (y >> 11) ^ y) & 1) << 10);
      tr[j] = sR[src];
      ti[j] = sI[src];
    }
    __syncthreads();
#pragma unroll
    for (int j = 0; j < 16; ++j) {
      const int y = tid + 256 * j;
      sR[y] = tr[j];
      sI[y] = ti[j];
    }
    __syncthreads();
  }

  // --- readout: out[b] = sum_y |psi_y|^2 * (sum_w sign_w(y) Wout[w]) + b ---
  float acc = 0.0f;
#pragma unroll
  for (int j = 0; j < 16; ++j) {
    const int y = tid + 256 * j;
    const float p = sR[y] * sR[y] + sI[y] * sI[y];
    float s = 0.0f;
#pragma unroll
    for (int w = 0; w < NQ; ++w) {
      const float ww = Wout[w];
      s += ((y >> (11 - w)) & 1) ? -ww : ww;
    }
    acc += p * s;
  }
  red[tid] = acc;
  __syncthreads();
  for (int st = 128; st > 0; st >>= 1) {
    if (tid < st) red[tid] += red[tid + st];
    __syncthreads();
  }
  if (tid == 0) out[b] = red[0] + bout[0];
}

// ---------------------------------------------------------------------------
extern "C" void kernel_launch(void* const* d_in, const int* in_sizes, int n_in,
                              void* d_out, int out_size, void* d_ws, size_t ws_size,
                              hipStream_t stream) {
  (void)in_sizes; (void)n_in; (void)out_size; (void)ws_size;
  const float* inputs = (const float*)d_in[0];   // [B, SEQ]
  const float* W_in   = (const float*)d_in[1];   // [NQ, SEQ]
  const float* q_w    = (const float*)d_in[2];   // [NQ, NL, 3]
  const float* W_out  = (const float*)d_in[3];   // [1, NQ]
  const float* b_out  = (const float*)d_in[4];   // [1]
  float* out = (float*)d_out;                    // [B, 1]

  float* gates = (float*)d_ws;                   // NL*3*2*256 = 12288 floats
  float* Wp    = gates + NL * 3 * 2 * 256;       // 16*64 = 1024 floats
  float* xbuf  = Wp + 16 * SEQL;                 // B*16 floats

  pad_kernel<<<1, 1024, 0, stream>>>(W_in, Wp);
  gate_kernel<<<NL * 3, 256, 0, stream>>>(q_w, gates);
  proj_kernel<<<BATCH / 16, 32, 0, stream>>>(inputs, Wp, xbuf);
  circuit_kernel<<<BATCH, 256, 0, stream>>>(xbuf, gates, W_out, b_out, out);
}